// GatV2_67448166416640
// MI455X (gfx1250) — compile-verified
//
#include <hip/hip_runtime.h>
#include <hip/hip_bf16.h>

#define FDIM 64

typedef float v2f __attribute__((ext_vector_type(2)));
typedef float v8f __attribute__((ext_vector_type(8)));

// ---------------------------------------------------------------------------
// Fused projection kernel:  h0 = x @ W_proj + b_proj ;  h = h0 @ W_gat
// One workgroup = 128 threads = 4 waves, handles a 16-row node block.
// Wave w computes column tile [16w, 16w+16) using V_WMMA_F32_16X16X4_F32.
// h0 block staged in LDS for the second GEMM.
//
// WMMA f32 16x16x4 fragment layout (wave32):
//   A (16x4):  lane l: m = l&15, kh = l>>4;  v0 = A[m][2*kh], v1 = A[m][2*kh+1]
//   B (4x16):  lane l: n = l&15, kh = l>>4;  v0 = B[2*kh][n], v1 = B[2*kh+1][n]
//   C/D (16x16): lane l: n = l&15; VGPR r holds M = r + 8*(l>>4)
// ---------------------------------------------------------------------------
__global__ __launch_bounds__(128) void gemm_proj_gat(
    const float* __restrict__ x,
    const float* __restrict__ Wp, const float* __restrict__ bp,
    const float* __restrict__ Wg,
    float* __restrict__ h0, float* __restrict__ h, int n)
{
    __shared__ float lds_h0[16][FDIM + 4];   // +4 pad to spread banks

    const int wave = threadIdx.x >> 5;       // 0..3 -> column tile
    const int lane = threadIdx.x & 31;
    const int m16  = lane & 15;              // row within tile (A), col (B/C)
    const int kh   = lane >> 4;              // K half-select
    const int row0 = blockIdx.x * 16;
    const int col0 = wave * 16;

    if (row0 + 16 > n) return;               // uniform per block: EXEC stays full

    const float* __restrict__ arow = x + (size_t)(row0 + m16) * FDIM;

    // ---- GEMM 1: h0 tile = x_block @ W_proj (+ b_proj) ----
    v8f c = {};
#pragma unroll
    for (int k0 = 0; k0 < FDIM; k0 += 4) {
        v2f a, b;
        a.x = arow[k0 + 2 * kh];
        a.y = arow[k0 + 2 * kh + 1];
        b.x = Wp[(size_t)(k0 + 2 * kh) * FDIM + col0 + m16];
        b.y = Wp[(size_t)(k0 + 2 * kh + 1) * FDIM + col0 + m16];
        c = __builtin_amdgcn_wmma_f32_16x16x4_f32(false, a, false, b,
                                                  (short)0, c, false, false);
    }
    const float bias = bp[col0 + m16];
#pragma unroll
    for (int r = 0; r < 8; ++r) {
        const int m = r + 8 * kh;
        const float v = c[r] + bias;
        lds_h0[m][col0 + m16] = v;
        h0[(size_t)(row0 + m) * FDIM + col0 + m16] = v;
    }
    __syncthreads();

    // ---- GEMM 2: h tile = h0_block @ W_gat (A-frags from LDS) ----
    v8f c2 = {};
#pragma unroll
    for (int k0 = 0; k0 < FDIM; k0 += 4) {
        v2f a, b;
        a.x = lds_h0[m16][k0 + 2 * kh];
        a.y = lds_h0[m16][k0 + 2 * kh + 1];
        b.x = Wg[(size_t)(k0 + 2 * kh) * FDIM + col0 + m16];
        b.y = Wg[(size_t)(k0 + 2 * kh + 1) * FDIM + col0 + m16];
        c2 = __builtin_amdgcn_wmma_f32_16x16x4_f32(false, a, false, b,
                                                   (short)0, c2, false, false);
    }
#pragma unroll
    for (int r = 0; r < 8; ++r) {
        const int m = r + 8 * kh;
        h[(size_t)(row0 + m) * FDIM + col0 + m16] = c2[r];
    }
}

// ---- ordered-uint encoding for float atomicMax (monotonic over all floats) --
__device__ __forceinline__ unsigned f2ord(float f) {
    unsigned u = __float_as_uint(f);
    return (u & 0x80000000u) ? ~u : (u | 0x80000000u);
}
__device__ __forceinline__ float ord2f(unsigned u) {
    return __uint_as_float((u & 0x80000000u) ? (u & 0x7FFFFFFFu) : ~u);
}

__device__ __forceinline__ float lrelu(float v) { return v > 0.f ? v : 0.2f * v; }

// ---------------------------------------------------------------------------
// Edge scores: e = a . LeakyReLU(h[src] + h[tgt]); segment max into emax[tgt].
// One thread per edge; rows are L2-resident (h = 25.6 MB << 192 MB L2).
// ---------------------------------------------------------------------------
__global__ __launch_bounds__(256) void edge_scores(
    const float* __restrict__ h, const int* __restrict__ src,
    const int* __restrict__ tgt, const float* __restrict__ a_gat,
    float* __restrict__ e_out, unsigned* __restrict__ emax, int E)
{
    const int e = blockIdx.x * blockDim.x + threadIdx.x;
    if (e >= E) return;
    const int s = src[e], t = tgt[e];
    const float4* __restrict__ hs = (const float4*)(h + (size_t)s * FDIM);
    const float4* __restrict__ ht = (const float4*)(h + (size_t)t * FDIM);
    const float4* __restrict__ av = (const float4*)a_gat;
    float acc = 0.f;
#pragma unroll
    for (int i = 0; i < FDIM / 4; ++i) {
        const float4 sv = hs[i], tv = ht[i], a = av[i];
        acc += a.x * lrelu(sv.x + tv.x);
        acc += a.y * lrelu(sv.y + tv.y);
        acc += a.z * lrelu(sv.z + tv.z);
        acc += a.w * lrelu(sv.w + tv.w);
    }
    e_out[e] = acc;
    atomicMax(&emax[t], f2ord(acc));
}

// ---------------------------------------------------------------------------
// e_exp = exp(e - emax[tgt]); denom[tgt] += e_exp
// ---------------------------------------------------------------------------
__global__ __launch_bounds__(256) void edge_expsum(
    const float* __restrict__ e_in, const int* __restrict__ tgt,
    const unsigned* __restrict__ emax, float* __restrict__ e_exp,
    float* __restrict__ denom, int E)
{
    const int e = blockIdx.x * blockDim.x + threadIdx.x;
    if (e >= E) return;
    const int t = tgt[e];
    const float ex = __expf(e_in[e] - ord2f(emax[t]));
    e_exp[e] = ex;
    atomicAdd(&denom[t], ex);
}

// ---------------------------------------------------------------------------
// agg[tgt] += alpha * h[src]; 4 threads per edge, 16 features each.
// ---------------------------------------------------------------------------
__global__ __launch_bounds__(256) void edge_scatter(
    const float* __restrict__ h, const float* __restrict__ e_exp,
    const float* __restrict__ denom, const int* __restrict__ src,
    const int* __restrict__ tgt, float* __restrict__ agg, int E)
{
    const int idx = blockIdx.x * blockDim.x + threadIdx.x;
    const int e = idx >> 2;
    if (e >= E) return;
    const int fo = (idx & 3) * 16;
    const int s = src[e], t = tgt[e];
    const float alpha = e_exp[e] / (denom[t] + 1e-16f);
    const float* __restrict__ hr = h + (size_t)s * FDIM + fo;
    float* __restrict__ ar = agg + (size_t)t * FDIM + fo;
#pragma unroll
    for (int i = 0; i < 16; ++i) atomicAdd(&ar[i], alpha * hr[i]);
}

// ---------------------------------------------------------------------------
// out = (agg + h0 + b_gat) @ W_pred + b_pred   (one thread per node)
// ---------------------------------------------------------------------------
__global__ __launch_bounds__(256) void pred_head(
    const float* __restrict__ agg, const float* __restrict__ h0,
    const float* __restrict__ bg, const float* __restrict__ Wpred,
    const float* __restrict__ bpred, float* __restrict__ out, int n)
{
    const int node = blockIdx.x * blockDim.x + threadIdx.x;
    if (node >= n) return;
    const float4* __restrict__ ag = (const float4*)(agg + (size_t)node * FDIM);
    const float4* __restrict__ hr = (const float4*)(h0 + (size_t)node * FDIM);
    const float4* __restrict__ bv = (const float4*)bg;
    const float4* __restrict__ wv = (const float4*)Wpred;
    float acc = 0.f;
#pragma unroll
    for (int i = 0; i < FDIM / 4; ++i) {
        const float4 a = ag[i], hh = hr[i], b = bv[i], w = wv[i];
        acc += (a.x + hh.x + b.x) * w.x;
        acc += (a.y + hh.y + b.y) * w.y;
        acc += (a.z + hh.z + b.z) * w.z;
        acc += (a.w + hh.w + b.w) * w.w;
    }
    out[node] = acc + bpred[0];
}

extern "C" void kernel_launch(void* const* d_in, const int* in_sizes, int n_in,
                              void* d_out, int out_size, void* d_ws, size_t ws_size,
                              hipStream_t stream)
{
    const float* x      = (const float*)d_in[0];
    const int*   adj    = (const int*)d_in[1];
    const float* W_proj = (const float*)d_in[2];
    const float* b_proj = (const float*)d_in[3];
    const float* W_gat  = (const float*)d_in[4];
    const float* a_gat  = (const float*)d_in[5];
    const float* b_gat  = (const float*)d_in[6];
    const float* W_pred = (const float*)d_in[7];
    const float* b_pred = (const float*)d_in[8];
    float* out = (float*)d_out;

    const int N = in_sizes[0] / FDIM;
    const int E = in_sizes[1] / 2;
    const int* src = adj;
    const int* tgt = adj + E;

    // workspace layout (16B-aligned slabs)
    char* ws = (char*)d_ws;
    size_t off = 0;
    float*    h0    = (float*)(ws + off); off += (size_t)N * FDIM * 4;
    float*    h     = (float*)(ws + off); off += (size_t)N * FDIM * 4;
    float*    agg   = (float*)(ws + off); off += (size_t)N * FDIM * 4;
    float*    e_buf = (float*)(ws + off); off += (size_t)E * 4;
    float*    e_exp = (float*)(ws + off); off += (size_t)E * 4;
    unsigned* emax  = (unsigned*)(ws + off); off += (size_t)N * 4;
    float*    denom = (float*)(ws + off); off += (size_t)N * 4;

    // zero accumulators (emax ordered-encoding: 0 == -NaN sentinel, < all reals;
    // emax and denom are adjacent -> one memset)
    hipMemsetAsync(agg, 0, (size_t)N * FDIM * 4, stream);
    hipMemsetAsync(emax, 0, (size_t)N * 8, stream);

    // 1) fused dual GEMM via f32 WMMA
    gemm_proj_gat<<<(N + 15) / 16, 128, 0, stream>>>(x, W_proj, b_proj, W_gat,
                                                     h0, h, N);
    // 2) attention logits + segment max
    edge_scores<<<(E + 255) / 256, 256, 0, stream>>>(h, src, tgt, a_gat,
                                                     e_buf, emax, E);
    // 3) exp + segment sum
    edge_expsum<<<(E + 255) / 256, 256, 0, stream>>>(e_buf, tgt, emax,
                                                     e_exp, denom, E);
    // 4) alpha-weighted scatter aggregation
    edge_scatter<<<((E * 4) + 255) / 256, 256, 0, stream>>>(h, e_exp, denom,
                                                            src, tgt, agg, E);
    // 5) skip + bias + prediction head
    pred_head<<<(N + 255) / 256, 256, 0, stream>>>(agg, h0, b_gat, W_pred,
                                                   b_pred, out, N);
}